// KernelPCA_12687333392561
// MI455X (gfx1250) — compile-verified
//
#include <hip/hip_runtime.h>

typedef __attribute__((ext_vector_type(2))) float v2f;
typedef __attribute__((ext_vector_type(8))) float v8f;

#define N_ROWS 2048
#define M_SV   1024
#define D_FEAT 256
#define C_OUT  128

#define BN   16     // x rows per workgroup
#define BM   16     // sv rows per tile
#define XPAD 268    // padded LDS row stride (floats): 268%64=12 -> conflict-free strided reads
#define KPAD 1028   // padded LDS row stride for K tile: 1028%64=4 -> conflict-free b64 frag loads

// Async global->LDS copy (gfx1250): per-lane 16B, tracked by ASYNCcnt.
__device__ __forceinline__ void async_copy_b128(unsigned lds_byte_addr, const void* gaddr) {
    asm volatile("global_load_async_to_lds_b128 %0, %1, off"
                 :: "v"(lds_byte_addr), "v"(gaddr)
                 : "memory");
}
#define ASYNC_WAIT(n) asm volatile("s_wait_asynccnt %0" :: "i"(n) : "memory")

// generic shared-memory pointer -> 32-bit LDS byte address (flat aperture: addr[31:0])
__device__ __forceinline__ unsigned lds_addr(const void* p) {
    return (unsigned)(unsigned long long)p;
}

__global__ __launch_bounds__(256) void kpca_fused(
    const float* __restrict__ x,
    const float* __restrict__ sv,
    const float* __restrict__ k_fit_rows,
    const float* __restrict__ k_fit_all_p,
    const float* __restrict__ alphas,
    float* __restrict__ out)
{
    __shared__ __align__(16) float xs[BN * XPAD];
    __shared__ __align__(16) float svs[2][BM * XPAD];   // double-buffered raw sv tiles
    __shared__ __align__(16) float Klds[BN * KPAD];
    __shared__ float sbuf[C_OUT];       // s[c] = colsum(alphas)
    __shared__ float bbuf[C_OUT];       // b[c] = k_fit_rows @ alphas
    __shared__ float rowpart[BN * 17];
    __shared__ float rowsum[BN];

    const int tid  = threadIdx.x;
    const int lane = tid & 31;
    const int wave = tid >> 5;
    const int row0 = blockIdx.x * BN;

    // a = sqrt(gamma * log2(e)), gamma = 1/256 ; exp(-g*(x-s)^2) == exp2(-(a*x - a*s)^2)
    const float gscale = 0.075070152f;
    const float ngs    = -gscale;

    // per-thread async-copy slots: chunks i = tid + k*256, i -> (row i>>6, quad i&63)
    const int cr0 = tid >> 6;            // rows serviced: cr0, cr0+4, cr0+8, cr0+12
    const int cq  = (tid & 63) * 4;      // float offset of 16B chunk within row

    // ---- Phase A: load + pre-scale x tile into LDS (synchronous, once) ----
    const float4* x4 = reinterpret_cast<const float4*>(x);
    for (int i = tid; i < BN * (D_FEAT / 4); i += 256) {
        int r = i >> 6, q = i & 63;
        float4 v = x4[(size_t)(row0 + r) * (D_FEAT / 4) + q];
        float* dst = &xs[r * XPAD + q * 4];
        dst[0] = v.x * gscale; dst[1] = v.y * gscale;
        dst[2] = v.z * gscale; dst[3] = v.w * gscale;
    }

    // ---- prologue: async-stage sv tile 0 into buffer 0 ----
    {
        #pragma unroll
        for (int k = 0; k < 4; ++k) {
            int r = cr0 + k * 4;
            async_copy_b128(lds_addr(&svs[0][r * XPAD + cq]),
                            &sv[(size_t)r * D_FEAT + cq]);
        }
    }

    // ---- Phase B: per-WG precompute of s[c], b[c] (folds centering into rank-1) ----
    if (tid < C_OUT) {
        float s0 = 0.f, b0 = 0.f;
        #pragma unroll 8
        for (int m = 0; m < M_SV; ++m) {
            float a  = alphas[(size_t)m * C_OUT + tid];
            float kr = k_fit_rows[m];
            s0 += a;
            b0 = __builtin_fmaf(kr, a, b0);
        }
        sbuf[tid] = s0; bbuf[tid] = b0;
    }

    // ---- Phase C: RBF-elementwise kernel matrix K[16][1024] into LDS ----
    const int nloc = tid >> 4;     // 0..15 : x row within tile
    const int mloc = tid & 15;     // 0..15 : sv row within tile
    float rsum = 0.f;

    for (int mt = 0; mt < M_SV / BM; ++mt) {
        const int cur = mt & 1;
        if (mt + 1 < M_SV / BM) {
            // async-stage next tile into the other buffer (overlaps with compute)
            #pragma unroll
            for (int k = 0; k < 4; ++k) {
                int r = cr0 + k * 4;
                async_copy_b128(lds_addr(&svs[cur ^ 1][r * XPAD + cq]),
                                &sv[(size_t)((mt + 1) * BM + r) * D_FEAT + cq]);
            }
            ASYNC_WAIT(4);   // older 4 copies (tile mt) complete; newer 4 in flight
        } else {
            ASYNC_WAIT(0);
        }
        __syncthreads();     // tile mt visible to all waves

        const float* xp = &xs[nloc * XPAD];
        const float* sp = &svs[cur][mloc * XPAD];
        float a0 = 0.f, a1 = 0.f, a2 = 0.f, a3 = 0.f;
        #pragma unroll 8
        for (int d = 0; d < D_FEAT; d += 4) {
            float4 xv = *reinterpret_cast<const float4*>(xp + d);
            float4 sw = *reinterpret_cast<const float4*>(sp + d);
            float t0 = __builtin_fmaf(sw.x, ngs, xv.x);
            float t1 = __builtin_fmaf(sw.y, ngs, xv.y);
            float t2 = __builtin_fmaf(sw.z, ngs, xv.z);
            float t3 = __builtin_fmaf(sw.w, ngs, xv.w);
            a0 += __builtin_amdgcn_exp2f(-(t0 * t0));
            a1 += __builtin_amdgcn_exp2f(-(t1 * t1));
            a2 += __builtin_amdgcn_exp2f(-(t2 * t2));
            a3 += __builtin_amdgcn_exp2f(-(t3 * t3));
        }
        float kv = (a0 + a1) + (a2 + a3);
        Klds[nloc * KPAD + mt * BM + mloc] = kv;
        rsum += kv;
        __syncthreads();     // all waves done reading buffer `cur` before it is re-filled
    }

    // ---- rowsum reduction (k_pred_cols = rowsum/1024) ----
    rowpart[nloc * 17 + mloc] = rsum;
    __syncthreads();
    if (tid < BN) {
        float s = 0.f;
        #pragma unroll
        for (int j = 0; j < 16; ++j) s += rowpart[tid * 17 + j];
        rowsum[tid] = s;
    }
    __syncthreads();

    // ---- Phase D: out(16x128) = K(16x1024) @ alphas(1024x128) via f32 WMMA ----
    const int c0  = wave * 16;     // 8 waves x 16 cols = 128
    const int l15 = lane & 15;
    const int lh  = lane >> 4;     // lane half selects K pair {0,1} vs {2,3}
    v8f acc = {};
    const float* Arow  = &Klds[l15 * KPAD];
    const float* Bbase = alphas + c0 + l15;

    #pragma unroll 8
    for (int k0 = 0; k0 < M_SV; k0 += 4) {
        int kk = k0 + lh * 2;
        v2f af; af.x = Arow[kk];                       // A: 16x4 f32 fragment (ds_load_2addr_b64)
                af.y = Arow[kk + 1];
        v2f bf; bf.x = Bbase[(size_t)kk * C_OUT];      // B: 4x16 f32 fragment
                bf.y = Bbase[(size_t)(kk + 1) * C_OUT];
        acc = __builtin_amdgcn_wmma_f32_16x16x4_f32(
            /*neg_a=*/false, af, /*neg_b=*/false, bf,
            /*c_mod=*/(short)0, acc, /*reuse_a=*/false, /*reuse_b=*/false);
    }

    // ---- epilogue: rank-1 centering correction + store ----
    const float kfa = k_fit_all_p[0];
    const float sc  = sbuf[c0 + l15];
    const float bc  = bbuf[c0 + l15];
    #pragma unroll
    for (int i = 0; i < 8; ++i) {
        int r = i + lh * 8;        // C/D layout: vgpr i -> row i (lanes 0-15) / i+8 (lanes 16-31)
        float corr = __builtin_fmaf(kfa - rowsum[r] * (1.0f / 1024.0f), sc, -bc);
        out[(size_t)(row0 + r) * C_OUT + c0 + l15] = acc[i] + corr;
    }
}

extern "C" void kernel_launch(void* const* d_in, const int* in_sizes, int n_in,
                              void* d_out, int out_size, void* d_ws, size_t ws_size,
                              hipStream_t stream) {
    (void)in_sizes; (void)n_in; (void)out_size; (void)d_ws; (void)ws_size;
    const float* x     = (const float*)d_in[0];
    const float* sv    = (const float*)d_in[1];
    const float* kfr   = (const float*)d_in[2];
    const float* kfa   = (const float*)d_in[3];
    const float* alph  = (const float*)d_in[4];
    float*       out   = (float*)d_out;

    kpca_fused<<<dim3(N_ROWS / BN), dim3(256), 0, stream>>>(x, sv, kfr, kfa, alph, out);
}